// get_intensity_histogram_10995116278400
// MI455X (gfx1250) — compile-verified
//
#include <hip/hip_runtime.h>
#include <stdint.h>

// -------- reference mapping -----------------------------------------------
// inputs : d_in[0] = batchsize (int, 1 elem), d_in[1] = input (float32, 8192*4096)
// output : d_out = [hist (256 f32) | count (256 f32)], count[i] = batchsize*hist[0]
// --------------------------------------------------------------------------

#define NBINS      256
#define TPB        256
#define NWAVES     (TPB / 32)       // wave32 on gfx1250
#define SUBSTRIDE  257              // odd stride -> per-wave LDS bank shift
#define VEC        4
#define MAXBLOCKS  2048

#if defined(__has_builtin)
#if __has_builtin(__builtin_amdgcn_global_load_async_to_lds_b128)
#define HAVE_ASYNC_LDS 1
#endif
#if __has_builtin(__builtin_amdgcn_s_wait_asynccnt)
#define HAVE_WAIT_ASYNC_BUILTIN 1
#endif
#endif

// Builtin signature (from clang diagnostic): params are pointers to
// int __attribute__((vector_size(16))) in AS1 (global src) / AS3 (LDS dst).
typedef int v4i __attribute__((__vector_size__(16)));
typedef __attribute__((address_space(1))) v4i gv4i;
typedef __attribute__((address_space(3))) v4i lv4i;

template <int N>
__device__ __forceinline__ void async_wait() {
#if defined(HAVE_WAIT_ASYNC_BUILTIN)
  __builtin_amdgcn_s_wait_asynccnt(N);
#else
  asm volatile("s_wait_asynccnt %0" :: "i"(N) : "memory");
#endif
}

__device__ __forceinline__ void bin_one(float v, unsigned int* h) {
  // faithful torch.histc semantics: width 1.0, [0,255], x==255 -> bin 255,
  // out-of-range dropped. v >= 0 so trunc == floor.
  if (v >= 0.0f && v <= 255.0f) {
    int b = (int)v;
    b = b > (NBINS - 1) ? (NBINS - 1) : b;
    atomicAdd(&h[b], 1u);              // ds_add_u32 (LDS atomic)
  }
}

__device__ __forceinline__ void bin_four(float4 v, unsigned int* h) {
  bin_one(v.x, h); bin_one(v.y, h); bin_one(v.z, h); bin_one(v.w, h);
}

__global__ void hist_init_kernel(unsigned int* __restrict__ gbins) {
  int t = threadIdx.x;
  if (t < NBINS) gbins[t] = 0u;
}

__global__ __launch_bounds__(TPB)
void hist_main_kernel(const float* __restrict__ in,
                      unsigned int* __restrict__ gbins,
                      long long n) {
  __shared__ unsigned int sub[NWAVES * SUBSTRIDE];
#if defined(HAVE_ASYNC_LDS)
  __shared__ __align__(16) float stage[2][TPB * VEC];   // 8 KB double buffer
#endif
  const int tid = threadIdx.x;
  const int wv  = tid >> 5;                 // wave32 id
  unsigned int* mysub = &sub[wv * SUBSTRIDE];

  for (int i = tid; i < NWAVES * SUBSTRIDE; i += TPB) sub[i] = 0u;
  __syncthreads();

  const long long stride = (long long)gridDim.x * TPB * VEC;
  const long long idx0   = ((long long)blockIdx.x * TPB + tid) * (long long)VEC;

#if defined(HAVE_ASYNC_LDS)
  // Double-buffered async global->LDS staging (ASYNCcnt pipeline).
  // Each lane stages its own 16B slot, so no cross-lane barrier is needed.
  if (idx0 + VEC <= n) {
    __builtin_amdgcn_global_load_async_to_lds_b128(
        (gv4i*)(in + idx0), (lv4i*)&stage[0][tid * VEC], 0, 0);
  }
  int buf = 0;
  for (long long i = idx0; i + VEC <= n; i += stride) {
    const long long nxt = i + stride;
    if (nxt + VEC <= n) {
      __builtin_amdgcn_global_load_async_to_lds_b128(
          (gv4i*)(in + nxt), (lv4i*)&stage[buf ^ 1][tid * VEC], 0, 0);
      async_wait<1>();   // async loads complete in order -> current buf ready
    } else {
      async_wait<0>();
    }
    float4 v = *(const float4*)&stage[buf][tid * VEC];  // ds_load_b128
    bin_four(v, mysub);
    buf ^= 1;
  }
#else
  for (long long i = idx0; i + VEC <= n; i += stride) {
    const long long nxt = i + stride;
    if (nxt + VEC <= n) __builtin_prefetch(in + nxt, 0, 0);  // global_prefetch_b8
    float4 v = *(const float4*)(in + i);                     // global_load_b128
    bin_four(v, mysub);
  }
#endif

  // scalar tail (n % VEC), handled by block 0 (empty for 33.5M elems)
  const long long nv = n & ~(long long)(VEC - 1);
  if (blockIdx.x == 0 && (long long)tid < (n - nv)) {
    bin_one(in[nv + tid], mysub);
  }

  __syncthreads();
  if (tid < NBINS) {
    unsigned int s = 0;
#pragma unroll
    for (int w = 0; w < NWAVES; ++w) s += sub[w * SUBSTRIDE + tid];
    if (s) atomicAdd(&gbins[tid], s);   // global_atomic_add_u32
  }
}

__global__ void hist_final_kernel(const unsigned int* __restrict__ gbins,
                                  const int* __restrict__ batchsize,
                                  float* __restrict__ out) {
  int t = threadIdx.x;
  if (t < NBINS) {
    out[t] = (float)gbins[t];
    out[NBINS + t] = (float)(*batchsize) * (float)gbins[0];
  }
}

extern "C" void kernel_launch(void* const* d_in, const int* in_sizes, int n_in,
                              void* d_out, int out_size, void* d_ws, size_t ws_size,
                              hipStream_t stream) {
  const int*   batchsize = (const int*)d_in[0];
  const float* input     = (const float*)d_in[1];
  const long long n      = (long long)in_sizes[1];

  unsigned int* gbins = (unsigned int*)d_ws;   // 256 * u32 scratch

  hist_init_kernel<<<1, NBINS, 0, stream>>>(gbins);

  long long tiles = (n + (long long)TPB * VEC - 1) / ((long long)TPB * VEC);
  int blocks = (int)(tiles < MAXBLOCKS ? (tiles > 0 ? tiles : 1) : MAXBLOCKS);
  hist_main_kernel<<<blocks, TPB, 0, stream>>>(input, gbins, n);

  hist_final_kernel<<<1, NBINS, 0, stream>>>(gbins, batchsize, (float*)d_out);
}